// TOPOLoss_window_13151189860984
// MI455X (gfx1250) — compile-verified
//
#include <hip/hip_runtime.h>
#include <stdint.h>

// ---------------------------------------------------------------------------
// MALIS loss for MI455X (gfx1250, wave32).
// Grid: 128 blocks (2 batches x 64 windows), 32 threads (1 wave) per block.
// Per block: TDM tensor_load_to_lds tile staging, bitmask dilation,
// min-label CCL in LDS, 2x (stable bitonic sort + segment-parallel Kruskal),
// node gather, WMMA f32 16x16x4 reduction of the per-pixel loss terms.
// d_ws layout: [0,512)   float partial[128]
//              [4096,..) u16 cnt pool, 64KB per block (128 * 65536 B).
// ---------------------------------------------------------------------------

typedef float v2f __attribute__((ext_vector_type(2)));
typedef float v8f __attribute__((ext_vector_type(8)));
typedef unsigned int u32x4 __attribute__((ext_vector_type(4)));
typedef int i32x8 __attribute__((ext_vector_type(8)));
typedef int i32x4 __attribute__((ext_vector_type(4)));

#define WIN 32
#define NODES 1024
#define EH 992          /* 32*31 horizontal edges */
#define NEDGE 1984
#define NSORT 2048

// Issue one TDM 2D tile load: 32x32 f32 tile from a 256-wide, 512-tall f32
// tensor (B*H x W view of [2,1,256,256]) into LDS at lds_off.
__device__ static inline void tdm_load_tile(const float* gptr, unsigned lds_off) {
  unsigned long long ga = (unsigned long long)(uintptr_t)gptr;
  u32x4 g0;
  g0[0] = 1u;                                        // count=1 (valid descriptor)
  g0[1] = lds_off;                                   // LDS byte address
  g0[2] = (unsigned)(ga & 0xFFFFFFFFull);            // global_addr[31:0]
  g0[3] = (unsigned)((ga >> 32) & 0x01FFFFFFull)     // global_addr[56:32]
          | (2u << 30);                              // type = 2 ("image")
  i32x8 g1;
  g1[0] = (int)(2u << 16);                           // data_size = 4 bytes
  g1[1] = (int)(256u << 16);                         // tensor_dim0[15:0] = 256
  g1[2] = (int)(512u << 16);                         // dim0 hi=0 | tensor_dim1 lo = 512
  g1[3] = (int)(32u << 16);                          // dim1 hi=0 | tile_dim0 = 32
  g1[4] = (int)(32u);                                // tile_dim1 = 32, tile_dim2 = 0
  g1[5] = (int)(256u);                               // tensor_dim0_stride lo32 = 256
  g1[6] = 0;                                         // stride0 hi16 | stride1[15:0]
  g1[7] = (int)((256u * 512u) >> 16);                // stride1[47:16]
  i32x4 z4; z4[0] = 0; z4[1] = 0; z4[2] = 0; z4[3] = 0;
  i32x8 z8; z8[0] = 0; z8[1] = 0; z8[2] = 0; z8[3] = 0;
  z8[4] = 0; z8[5] = 0; z8[6] = 0; z8[7] = 0;
  // clang-23 / therock-10.0 signature: (u32x4, i32x8, i32x4, i32x4, i32x8, i32 cpol)
  __builtin_amdgcn_tensor_load_to_lds(g0, g1, z4, z4, z8, 0);
}

__device__ static inline void edge_ab(int e, int& a, int& b) {
  if (e < EH) { int r = e / 31; int c = e - r * 31; a = r * 32 + c; b = a + 1; }
  else        { int t = e - EH; a = t; b = t + 32; }
}

// Uniform union-find find with full path compression. All lanes execute with identical
// data; duplicate identical LDS stores are benign.
__device__ static inline int uf_find(volatile unsigned short* par, int x) {
  int r = x;
  while (par[r] != r) r = par[r];
  while (par[x] != (unsigned short)r) {
    int p = par[x];
    par[x] = (unsigned short)r;
    x = p;
  }
  return r;
}

__global__ __launch_bounds__(32)
void malis_window_kernel(const float* __restrict__ pred,
                         const float* __restrict__ targ,
                         const float* __restrict__ lr_p,
                         const float* __restrict__ lrp_p,
                         float* __restrict__ partial,
                         unsigned short* __restrict__ cntpool) {
  __shared__ float predT[NODES];
  __shared__ float targT[NODES];
  __shared__ unsigned short lab[NODES];
  __shared__ unsigned short seg[NODES];
  __shared__ unsigned short par[NODES];
  __shared__ unsigned long long keys[NSORT];
  __shared__ float w[NEDGE];
  __shared__ float lossbuf[NODES];
  __shared__ unsigned short seglist[32];
  __shared__ int flag;

  const int tid = threadIdx.x;        // 0..31 == tile row
  const int blk = blockIdx.x;         // 0..127
  const int bimg = blk >> 6;          // batch index
  const int win  = blk & 63;
  const int wr = win >> 3;
  const int wc = win & 7;
  const float lr  = lr_p[0];
  const float lrp = lrp_p[0];

  unsigned short* cnt = cntpool + (size_t)blk * (size_t)(NODES * 32);

  // ---- TDM async tensor loads of the two 32x32 tiles into LDS ----
  const size_t base = ((size_t)bimg * 256 + (size_t)wr * 32) * 256 + (size_t)wc * 32;
  tdm_load_tile(pred + base, (unsigned)(uintptr_t)(void*)predT);
  tdm_load_tile(targ + base, (unsigned)(uintptr_t)(void*)targT);
  __builtin_amdgcn_s_wait_tensorcnt(0);
  __syncthreads();

  // ---- mask = (t == 0) as one 32-bit row mask per lane; dilate 5x (4-conn) ----
  unsigned m = 0u;
  for (int c = 0; c < 32; ++c)
    if (targT[tid * 32 + c] == 0.0f) m |= (1u << c);
  for (int it = 0; it < 5; ++it) {
    unsigned up = __shfl_up(m, 1, 32);   if (tid == 0)  up = 0u;
    unsigned dn = __shfl_down(m, 1, 32); if (tid == 31) dn = 0u;
    m = m | (m << 1) | (m >> 1) | up | dn;
  }
  const unsigned validbits = ~m;   // mask2 = (dilated == 0)

  // ---- init labels: valid -> i+1, invalid -> 0xFFFF sentinel ----
  for (int c = 0; c < 32; ++c) {
    int i = tid * 32 + c;
    lab[i] = ((validbits >> c) & 1u) ? (unsigned short)(i + 1) : (unsigned short)0xFFFF;
  }
  __syncthreads();

  // ---- 8-connected min-label propagation to fixpoint ----
  {
    volatile unsigned short* vlab = lab;
    for (;;) {
      if (tid == 0) flag = 0;
      __syncthreads();
      int changed = 0;
      for (int c = 0; c < 32; ++c) {
        int i = tid * 32 + c;
        int cur = vlab[i];
        if (cur == 0xFFFF) continue;
        int mn = cur;
        for (int dr = -1; dr <= 1; ++dr) {
          int rr = tid + dr;
          if (rr < 0 || rr > 31) continue;
          for (int dc = -1; dc <= 1; ++dc) {
            int cc = c + dc;
            if (cc < 0 || cc > 31) continue;
            int v = vlab[rr * 32 + cc];
            if (v < mn) mn = v;
          }
        }
        if (mn < cur) { vlab[i] = (unsigned short)mn; changed = 1; }
      }
      if (changed) flag = 1;
      __syncthreads();
      if (!flag) break;
    }
  }

  // ---- map distinct labels -> segment ids 1..31 (serial on lane 0) ----
  if (tid == 0) {
    int ns = 0;
    for (int i = 0; i < NODES; ++i) {
      unsigned short v = lab[i];
      if (v == 0xFFFF) { seg[i] = 0; continue; }
      int s = -1;
      for (int k2 = 0; k2 < ns; ++k2)
        if (seglist[k2] == v) { s = k2; break; }
      if (s < 0) {
        if (ns < 31) { seglist[ns] = v; s = ns; ++ns; }
        else s = 31;                       // overflow bucket (counts in totals only)
      }
      seg[i] = (unsigned short)(s + 1);
    }
  }
  __syncthreads();

  // ======================= two MALIS passes =======================
  for (int pass = 0; pass < 2; ++pass) {
    const bool pos = (pass == 1);

    // ---- build stable-descending sort keys ----
    for (int e = tid; e < NSORT; e += 32) {
      unsigned long long kk;
      if (e < NEDGE) {
        int a, b; edge_ab(e, a, b);
        float cost = predT[a] + predT[b];
        float g    = targT[a] + targT[b];
        if (pos) { if (g < 10.0f) cost = 0.0f; }
        else     { if (g > 20.0f) cost = 20.0f; }
        unsigned u = __float_as_uint(cost);
        u ^= (u >> 31) ? 0xFFFFFFFFu : 0x80000000u;   // total-order ascending map
        kk = ((unsigned long long)(~u) << 32) | (unsigned)e; // desc cost, asc index
      } else {
        kk = ~0ull;                                   // pad sorts last
      }
      keys[e] = kk;
    }
    for (int i = tid; i < NODES; i += 32) par[i] = (unsigned short)i;
    for (int e = tid; e < NEDGE; e += 32) w[e] = 0.0f;
    // per-lane segment counts: lane 0 = total labeled, lane s = segment s
    for (int i = 0; i < NODES; ++i) {
      int s = seg[i];
      unsigned short v = (tid == 0) ? (unsigned short)(s != 0)
                                    : (unsigned short)(s == tid);
      cnt[i * 32 + tid] = v;
    }
    __syncthreads();

    // ---- bitonic sort (ascending keys == descending cost, stable) ----
    for (unsigned k2 = 2; k2 <= NSORT; k2 <<= 1) {
      for (unsigned j = k2 >> 1; j > 0; j >>= 1) {
        for (unsigned i = tid; i < NSORT; i += 32) {
          unsigned l2 = i ^ j;
          if (l2 > i) {
            unsigned long long x = keys[i], y = keys[l2];
            bool asc = ((i & k2) == 0);
            if ((x > y) == asc) { keys[i] = y; keys[l2] = x; }
          }
        }
        __syncthreads();
      }
    }

    // ---- Kruskal max-spanning forest; lanes carry per-segment counts ----
    for (int ei = 0; ei < NEDGE; ++ei) {
      int e = (int)(unsigned)(keys[ei] & 0xFFFFFFFFull);
      int a, b; edge_ab(e, a, b);
      int ra = uf_find(par, a);
      int rb = uf_find(par, b);
      if (ra == rb) continue;
      int ca = cnt[ra * 32 + tid];
      int cb = cnt[rb * 32 + tid];
      float prod = (float)ca * (float)cb;
      float same = (tid == 0) ? 0.0f : prod;
      for (int off = 16; off; off >>= 1) same += __shfl_xor(same, off, 32);
      float tot = __shfl(prod, 0, 32);               // sa * sb
      if (tid == 0) {
        w[e] = pos ? same : (tot - same);
        par[rb] = (unsigned short)ra;
      }
      cnt[ra * 32 + tid] = (unsigned short)(ca + cb);
    }
    __syncthreads();

    // ---- normalization sum (integer-valued, exact in f32) ----
    float sn = 0.0f;
    for (int e = tid; e < NEDGE; e += 32) sn += w[e];
    for (int off = 16; off; off >>= 1) sn += __shfl_xor(sn, off, 32);

    // masked, normalized edge weight
    auto edge_w = [&](int e) -> float {
      int a, b; edge_ab(e, a, b);
      float g = targT[a] + targT[b];
      bool z = pos ? (g < 20.0f) : (g >= 10.0f);
      if (z) return 0.0f;
      float v = w[e];
      return (sn > 0.0f) ? (v / sn) : v;
    };

    // ---- gather edge weights to nodes, accumulate loss terms ----
    for (int i = tid; i < NODES; i += 32) {
      int r = i >> 5, c = i & 31;
      float nv = 0.0f;
      if (c <= 30) nv += edge_w(r * 31 + c);            // right  (nodes[:, :-1] += wh)
      if (c >= 1)  nv += edge_w(r * 31 + (c - 1));      // left   (nodes[:, 1:]  += wh)
      if (r <= 30) nv += edge_w(EH + r * 32 + c);       // down   (nodes[:-1, :] += wv)
      if (r >= 1)  nv += edge_w(EH + (r - 1) * 32 + c); // up     (nodes[1:, :]  += wv)
      float p = predT[i];
      float term = pos ? (lrp * (20.0f - p) * (20.0f - p) * nv)
                       : (lr  * p * p * nv);
      if (pass == 0) lossbuf[i] = term;
      else           lossbuf[i] += term;
    }
    __syncthreads();
  }

  // ---- WMMA f32 16x16x4 reduction of the 1024 loss terms ----
  // C += A(64 vals) x B(ones): accumulator row M holds running sum of that
  // row's slice, replicated over N. Total = sum_M D[M][0].
  v8f acc = {0.0f, 0.0f, 0.0f, 0.0f, 0.0f, 0.0f, 0.0f, 0.0f};
  v2f ones; ones.x = 1.0f; ones.y = 1.0f;
  for (int ch = 0; ch < 16; ++ch) {
    v2f a2;
    a2.x = lossbuf[ch * 64 + 2 * tid];
    a2.y = lossbuf[ch * 64 + 2 * tid + 1];
    acc = __builtin_amdgcn_wmma_f32_16x16x4_f32(false, a2, false, ones,
                                                (short)0, acc, false, false);
  }
  float ssum = 0.0f;
#pragma unroll
  for (int v = 0; v < 8; ++v) ssum += acc[v];
  float other_half = __shfl(ssum, 16, 32);  // lane0: M=0..7, lane16: M=8..15 (N=0)
  if (tid == 0) partial[blk] = ssum + other_half;
}

__global__ __launch_bounds__(32)
void reduce_partials_kernel(const float* __restrict__ partial,
                            float* __restrict__ out) {
  int t = threadIdx.x;
  float s = partial[t] + partial[t + 32] + partial[t + 64] + partial[t + 96];
  for (int off = 16; off; off >>= 1) s += __shfl_xor(s, off, 32);
  if (t == 0) out[0] = s;
}

extern "C" void kernel_launch(void* const* d_in, const int* in_sizes, int n_in,
                              void* d_out, int out_size, void* d_ws, size_t ws_size,
                              hipStream_t stream) {
  (void)in_sizes; (void)n_in; (void)out_size; (void)ws_size;
  const float* pred = (const float*)d_in[0];
  const float* targ = (const float*)d_in[1];
  const float* lr   = (const float*)d_in[2];
  const float* lrp  = (const float*)d_in[3];
  float* partial = (float*)d_ws;                                   // 128 floats
  unsigned short* cntpool = (unsigned short*)((char*)d_ws + 4096); // 128 * 64KB
  malis_window_kernel<<<dim3(128), dim3(32), 0, stream>>>(
      pred, targ, lr, lrp, partial, cntpool);
  reduce_partials_kernel<<<dim3(1), dim3(32), 0, stream>>>(
      partial, (float*)d_out);
}